// SimpleRoIAlign_38551626449367
// MI455X (gfx1250) — compile-verified
//
#include <hip/hip_runtime.h>

// Problem constants from the reference: B=2, C=256, H=200, W=336, R=1000
// out = (R, C, 14, 14) fp32 ; spatial_scale = 0.25
constexpr int OUT_H = 14, OUT_W = 14, OUT_HW = OUT_H * OUT_W;
constexpr int C_ = 256, H_ = 200, W_ = 336;
constexpr float SPATIAL_SCALE = 0.25f;

constexpr int CPB     = 32;   // channels per block -> grid.y = 256/32 = 8
constexpr int THREADS = 224;  // 7 wave32s; lanes 196..223 are masked out

__global__ __launch_bounds__(THREADS) void roi_align_kernel(
    const float* __restrict__ feat,
    const float* __restrict__ rois,
    float* __restrict__ out)
{
    const int r  = blockIdx.x;
    const int c0 = blockIdx.y * CPB;
    const int s  = threadIdx.x;
    const bool active = (s < OUT_HW);

    // RoI params are uniform per block (blockIdx-derived address) -> s_load path
    const float bf = rois[r * 5 + 0];
    const float x1 = rois[r * 5 + 1];
    const float y1 = rois[r * 5 + 2];
    const float x2 = rois[r * 5 + 3];
    const float y2 = rois[r * 5 + 4];
    const int   b  = (int)bf;

    const int oh = s / OUT_W;          // garbage for s>=196, masked below
    const int ow = s - oh * OUT_W;

    // fx = (x1 + (ow+0.5)/14 * (x2-x1)) * scale - 0.5   (matches reference)
    const float fx = (x1 + (ow + 0.5f) * (1.0f / OUT_W) * (x2 - x1)) * SPATIAL_SCALE - 0.5f;
    const float fy = (y1 + (oh + 0.5f) * (1.0f / OUT_H) * (y2 - y1)) * SPATIAL_SCALE - 0.5f;

    const float x0f = floorf(fx), y0f = floorf(fy);
    const float lx  = fx - x0f,   ly  = fy - y0f;
    const int x0  = (int)x0f, y0  = (int)y0f;
    const int x1i = x0 + 1,   y1i = y0 + 1;

    // clipped indices for the load, validity folded into the weights
    const int cx0 = min(max(x0,  0), W_ - 1);
    const int cx1 = min(max(x1i, 0), W_ - 1);
    const int cy0 = min(max(y0,  0), H_ - 1);
    const int cy1 = min(max(y1i, 0), H_ - 1);

    const bool vx0 = (x0  >= 0) & (x0  < W_);
    const bool vx1 = (x1i >= 0) & (x1i < W_);
    const bool vy0 = (y0  >= 0) & (y0  < H_);
    const bool vy1 = (y1i >= 0) & (y1i < H_);

    float w00 = (vx0 & vy0) ? (1.0f - lx) * (1.0f - ly) : 0.0f;
    float w01 = (vx1 & vy0) ? lx * (1.0f - ly)          : 0.0f;
    float w10 = (vx0 & vy1) ? (1.0f - lx) * ly          : 0.0f;
    float w11 = (vx1 & vy1) ? lx * ly                   : 0.0f;

    int i00 = cy0 * W_ + cx0;
    int i01 = cy0 * W_ + cx1;
    int i10 = cy1 * W_ + cx0;
    int i11 = cy1 * W_ + cx1;

    if (!active) {               // idle lanes: safe addresses, zero weights
        i00 = i01 = i10 = i11 = 0;
        w00 = w01 = w10 = w11 = 0.0f;
    }

    const size_t HW   = (size_t)H_ * W_;
    const float* base = feat + ((size_t)b * C_ + c0) * HW;
    float*       ob   = out  + ((size_t)r * C_ + c0) * OUT_HW + s;

    #pragma unroll 4
    for (int c = 0; c < CPB; ++c) {
        const float* p = base + (size_t)c * HW;

        // CDNA5 global_prefetch_b8: pull channel c+4's plane corner
        // lines toward the WGP (planes are ~268 KB apart; L2-resident data,
        // this hides the L2->L0 hop). Speculative, no counter cost.
        if (c + 4 < CPB) {
            __builtin_prefetch(p + 4 * HW + i00, 0, 1);
            __builtin_prefetch(p + 4 * HW + i11, 0, 1);
        }

        float v = w00 * p[i00];
        v = fmaf(w01, p[i01], v);
        v = fmaf(w10, p[i10], v);
        v = fmaf(w11, p[i11], v);

        // Non-temporal store: 200 MB output stream must NOT evict the
        // 137 MB feature tensor from the 192 MB L2.
        if (active) __builtin_nontemporal_store(v, ob + c * OUT_HW);
    }
}

extern "C" void kernel_launch(void* const* d_in, const int* in_sizes, int n_in,
                              void* d_out, int out_size, void* d_ws, size_t ws_size,
                              hipStream_t stream) {
    const float* feat = (const float*)d_in[0];
    const float* rois = (const float*)d_in[1];
    float*       out  = (float*)d_out;

    const int R = in_sizes[1] / 5;        // 1000
    dim3 grid(R, C_ / CPB);               // (1000, 8) blocks
    roi_align_kernel<<<grid, THREADS, 0, stream>>>(feat, rois, out);
}